// Model_14834817040511
// MI455X (gfx1250) — compile-verified
//
#include <hip/hip_runtime.h>

typedef __attribute__((ext_vector_type(2))) float v2f;
typedef __attribute__((ext_vector_type(8))) float v8f;

namespace {
constexpr int T = 1024, B = 512, D = 16, H = 16;

// CDNA5 hardware tanh (TRANS op). ISA §7.4.3: one independent op / V_NOP after
// a TRANS op before its output is consumed -> include v_nop in the asm.
__device__ __forceinline__ float tanh_hw(float x) {
  float r;
  asm volatile("v_tanh_f32 %0, %1\n\tv_nop" : "=v"(r) : "v"(x));
  return r;
}
__device__ __forceinline__ float sigmoid_hw(float x) {
  return __builtin_fmaf(0.5f, tanh_hw(0.5f * x), 0.5f);
}
__device__ __forceinline__ v8f splat8(float s) {
  v8f v;
#pragma unroll
  for (int i = 0; i < 8; ++i) v[i] = s;
  return v;
}

// Load the x tile for timestep t directly in WMMA A-layout:
// lane holds batch row n = lane&15, K-step j covers k = 4j + 2*hi + {0,1}.
__device__ __forceinline__ void load_x(v2f (&xa)[4], const float* __restrict__ x,
                                       int t, int bbase, int n, int hi) {
#pragma unroll
  for (int j = 0; j < 4; ++j)
    xa[j] = *(const v2f*)(x + ((size_t)t * B + bbase + n) * D + 4 * j + 2 * hi);
}

// One full LSTM timestep. G starts from the persistent bias splat passed as
// the C operand of the first WMMA per gate tile (no accumulator-init movs).
__device__ __forceinline__ void lstm_step(
    int t, const v2f (&xc)[4], const v2f (&wih)[4][4], const v2f (&whh)[4][4],
    const v8f (&biasV)[4], v2f (&hA)[4], v8f& c, float* __restrict__ y,
    float* hlds, int n, int hi, int bbase) {
  v8f G[4];
#pragma unroll
  for (int nt = 0; nt < 4; ++nt) {
    G[nt] = __builtin_amdgcn_wmma_f32_16x16x4_f32(
        false, xc[0], false, wih[nt][0], (short)0, biasV[nt], false, false);
#pragma unroll
    for (int j = 1; j < 4; ++j)
      G[nt] = __builtin_amdgcn_wmma_f32_16x16x4_f32(
          false, xc[j], false, wih[nt][j], (short)0, G[nt], false, false);
#pragma unroll
    for (int j = 0; j < 4; ++j)
      G[nt] = __builtin_amdgcn_wmma_f32_16x16x4_f32(
          false, hA[j], false, whh[nt][j], (short)0, G[nt], false, false);
  }

  // Gates (i,f,g,o = nt 0..3), cell update, output — all in C/D layout
  // (M = r + 8*hi, N = n).
  const int m0 = 8 * hi;
  float* yrow = y + ((size_t)t * B + bbase + m0) * H + n;
#pragma unroll
  for (int r = 0; r < 8; ++r) {
    float iv = sigmoid_hw(G[0][r]);
    float fv = sigmoid_hw(G[1][r]);
    float gv = tanh_hw(G[2][r]);
    float ov = sigmoid_hw(G[3][r]);
    float cv = __builtin_fmaf(fv, c[r], iv * gv);
    c[r] = cv;
    float hv = ov * tanh_hw(cv);
    yrow[(size_t)r * H] = hv;           // y[t][bbase+m0+r][n]
    hlds[(m0 + r) * 16 + n] = hv;       // stage h[b][k] for lane transpose
  }
  __syncthreads();                      // single wave: dscnt wait + nop
#pragma unroll
  for (int j = 0; j < 4; ++j)           // reload h in A-layout
    hA[j] = *(const v2f*)(hlds + n * 16 + 4 * j + 2 * hi);
  __syncthreads();
}
} // namespace

// One wave per 16-row batch tile (32 independent waves across WGPs; the only
// true serial dependency is the T=1024 recurrence). Weights, cell state and
// bias accumulator seeds are fully register-resident; per timestep:
// 32x v_wmma_f32_16x16x4_f32 + hardware v_tanh gates + one LDS round-trip.
__global__ __launch_bounds__(32, 1) void lstm_wmma_kernel(
    const float* __restrict__ x, const float* __restrict__ W_ih,
    const float* __restrict__ W_hh, const float* __restrict__ b_ih,
    const float* __restrict__ b_hh, float* __restrict__ y) {
  __shared__ __align__(16) float hlds[16 * 16];

  const int lane  = threadIdx.x;   // 0..31, wave32
  const int n     = lane & 15;
  const int hi    = lane >> 4;
  const int bbase = blockIdx.x * 16;

  // B-operand layout (4x16, K-major mirror of the documented A 16x4 layout):
  // lanes 0-15 hold N=lane, VGPR0/1 = K0/K1; lanes 16-31 hold K2/K3.
  // B[k][nn] = W[16*nt + nn][k]  (both GEMMs multiply by W^T).
  v2f wih[4][4], whh[4][4];
#pragma unroll
  for (int nt = 0; nt < 4; ++nt)
#pragma unroll
    for (int j = 0; j < 4; ++j) {
      wih[nt][j] = *(const v2f*)(W_ih + (size_t)(16 * nt + n) * D + 4 * j + 2 * hi);
      whh[nt][j] = *(const v2f*)(W_hh + (size_t)(16 * nt + n) * H + 4 * j + 2 * hi);
    }

  // Persistent bias splats, used as the C operand of each tile's first WMMA.
  v8f biasV[4];
#pragma unroll
  for (int nt = 0; nt < 4; ++nt)
    biasV[nt] = splat8(b_ih[16 * nt + n] + b_hh[16 * nt + n]);

  v8f c = {};                       // cell state, C/D layout
  v2f hA[4];                        // h in A-layout (zero initial state)
#pragma unroll
  for (int j = 0; j < 4; ++j) { hA[j][0] = 0.f; hA[j][1] = 0.f; }

  // Two x A-operand buffers, alternated by a manually 2x-unrolled time loop
  // so prefetched loads land directly in the consuming buffer (no reg copies).
  v2f xA0[4], xA1[4];
  load_x(xA0, x, 0, bbase, n, hi);

  for (int t = 0; t < T; t += 2) {
    load_x(xA1, x, t + 1, bbase, n, hi);          // prefetch for step t+1
    if (t + 16 < T)                               // deep prefetch (uniform)
      __builtin_prefetch(x + ((size_t)(t + 16) * B + bbase + n) * D, 0, 1);
    lstm_step(t, xA0, wih, whh, biasV, hA, c, y, hlds, n, hi, bbase);

    if (t + 2 < T) load_x(xA0, x, t + 2, bbase, n, hi);  // prefetch for t+2
    lstm_step(t + 1, xA1, wih, whh, biasV, hA, c, y, hlds, n, hi, bbase);
  }
}

extern "C" void kernel_launch(void* const* d_in, const int* /*in_sizes*/,
                              int /*n_in*/, void* d_out, int /*out_size*/,
                              void* /*d_ws*/, size_t /*ws_size*/,
                              hipStream_t stream) {
  const float* x    = (const float*)d_in[0];
  const float* W_ih = (const float*)d_in[1];
  const float* W_hh = (const float*)d_in[2];
  const float* b_ih = (const float*)d_in[3];
  const float* b_hh = (const float*)d_in[4];
  float* y = (float*)d_out;
  lstm_wmma_kernel<<<B / 16, 32, 0, stream>>>(x, W_ih, W_hh, b_ih, b_hh, y);
}